// DEIMTransformer_48086453846033
// MI455X (gfx1250) — compile-verified
//
#include <hip/hip_runtime.h>
#include <hip/hip_bf16.h>
#include <math.h>

// ---------------------------------------------------------------------------
// Problem constants (from the reference)
// ---------------------------------------------------------------------------
#define BB 16
#define QQ 300
#define CC 256
#define NH 8
#define NP 4
#define NL 4
#define HD 32          // CC / NH
#define SS 10500       // sum of h*w over levels

typedef __attribute__((ext_vector_type(2))) float v2f;
typedef __attribute__((ext_vector_type(8))) float v8f;

// ---------------------------------------------------------------------------
// Kernel 1: fp32 WMMA GEMM with bias:  Out[M,N] = X[M,K] * W[K,N] + bias[N]
// N, K are compile-time so ALL load addresses are immediate offsets from two
// base pointers (no per-iteration 64-bit address math). One wave per 16x32
// output tile (two 16x16 WMMA tiles sharing the A fragment), K stepped by 4
// with V_WMMA_F32_16X16X4_F32, fully unrolled.
//
// Fragment layouts (CDNA5 ISA 7.12.2, 32-bit data, wave32):
//   A (16x4):  lanes 0-15 -> M=lane,  V0=K0, V1=K1 ; lanes 16-31 -> V0=K2, V1=K3
//   B (4x16):  lanes 0-15 -> N=lane,  V0=K0, V1=K1 ; lanes 16-31 -> V0=K2, V1=K3
//   C/D (16x16): VGPR r: lanes 0-15 -> M=r, N=lane ; lanes 16-31 -> M=r+8
// ---------------------------------------------------------------------------
template <int N, int K>
__global__ void gemm_wmma_f32_bias(const float* __restrict__ X,
                                   const float* __restrict__ W,
                                   const float* __restrict__ bias,
                                   float* __restrict__ Out,
                                   int M) {
    constexpr int TILES_N = N / 32;                 // 16x32 macro-tiles per row
    const int wavesPerBlock = blockDim.x >> 5;
    const int wave = blockIdx.x * wavesPerBlock + (threadIdx.x >> 5);
    const int lane = threadIdx.x & 31;

    const int totalTiles = (M >> 4) * TILES_N;
    if (wave >= totalTiles) return;

    const int tm = wave / TILES_N;
    const int tn = wave - tm * TILES_N;

    const int halfLane = lane & 15;
    const int kOff = (lane >> 4) << 1;              // 0 (lanes 0-15) or 2 (16-31)

    // base pointers; every subsequent access is an immediate offset
    const float* __restrict__ xrow = X + (size_t)(tm * 16 + halfLane) * K + kOff;
    const int col0 = tn * 32 + halfLane;
    const float* __restrict__ wcol = W + (size_t)kOff * N + col0;

    v8f acc0 = {};
    v8f acc1 = {};

    #pragma unroll
    for (int k = 0; k < K; k += 4) {
        v2f a, b0, b1;
        a.x  = xrow[k];                 // X[row][k+kOff]     (b64 pair with .y)
        a.y  = xrow[k + 1];             // X[row][k+kOff+1]
        b0.x = wcol[k * N];             // W[k+kOff  ][col0]
        b0.y = wcol[k * N + N];         // W[k+kOff+1][col0]
        b1.x = wcol[k * N + 16];        // W[k+kOff  ][col0+16]
        b1.y = wcol[k * N + N + 16];    // W[k+kOff+1][col0+16]
        acc0 = __builtin_amdgcn_wmma_f32_16x16x4_f32(
                   false, a, false, b0, (short)0, acc0, false, false);
        acc1 = __builtin_amdgcn_wmma_f32_16x16x4_f32(
                   false, a, false, b1, (short)0, acc1, false, false);
    }

    // Write D (+bias). Each half-wave stores 64B contiguous per row per tile.
    const int nOut  = tn * 32 + halfLane;
    const int mBase = tm * 16 + ((lane < 16) ? 0 : 8);
    const float bv0 = bias[nOut];
    const float bv1 = bias[nOut + 16];
    #pragma unroll
    for (int r = 0; r < 8; ++r) {
        Out[(size_t)(mBase + r) * N + nOut]      = acc0[r] + bv0;
        Out[(size_t)(mBase + r) * N + nOut + 16] = acc1[r] + bv1;
    }
}

// ---------------------------------------------------------------------------
// Kernel 2: deformable-attention sampling + softmax + aggregation.
// One wave32 per (b, q, head); lane = channel within the head (HD == 32).
// Sampling locations are wave-uniform -> validity branches are uniform and
// each corner load is one coalesced 128B transaction from the L2-resident
// value tensor (172 MB < 192 MB global L2).
// ---------------------------------------------------------------------------
__global__ void msda_sample_agg(const float* __restrict__ memory,      // [B,S,C]
                                const float* __restrict__ ref_points,  // [B,Q,NL,2]
                                const float* __restrict__ offs,        // [B,Q,NH,NL,NP,2]
                                const float* __restrict__ logits,      // [B,Q,NH,NL*NP]
                                float* __restrict__ attended) {        // [B,Q,C]
    constexpr int lvl_h[NL]   = {80, 40, 20, 30};
    constexpr int lvl_w[NL]   = {80, 40, 20, 70};
    constexpr int lvl_off[NL] = {0, 6400, 8000, 8400};

    const int wavesPerBlock = blockDim.x >> 5;
    const int wave = blockIdx.x * wavesPerBlock + (threadIdx.x >> 5);
    const int lane = threadIdx.x & 31;

    const int total = BB * QQ * NH;
    if (wave >= total) return;

    const int h  = wave % NH;
    const int bq = wave / NH;         // b*Q + q
    const int b  = bq / QQ;

    // ---- softmax over the 16 (level, point) logits for this (b,q,h) ----
    float la[NL * NP];
    const float* lp = logits + ((size_t)bq * NH + h) * (NL * NP);
    float m = -1e30f;
    #pragma unroll
    for (int i = 0; i < NL * NP; ++i) { la[i] = lp[i]; m = fmaxf(m, la[i]); }
    float ssum = 0.f;
    #pragma unroll
    for (int i = 0; i < NL * NP; ++i) { la[i] = __expf(la[i] - m); ssum += la[i]; }
    const float inv = 1.0f / ssum;

    const float* __restrict__ offp = offs + (((size_t)bq * NH + h) * NL) * (NP * 2);
    const float* __restrict__ refp = ref_points + (size_t)bq * NL * 2;
    // base pointer for this (b, head, lane-channel); gather stride is C floats
    const float* __restrict__ memb = memory + (size_t)b * SS * CC + h * HD + lane;

    float acc = 0.f;

    #pragma unroll
    for (int l = 0; l < NL; ++l) {
        const int   hh   = lvl_h[l];
        const int   ww   = lvl_w[l];
        const int   base = lvl_off[l];
        const float fw   = (float)ww;
        const float fh   = (float)hh;
        const float rx   = refp[l * 2 + 0];
        const float ry   = refp[l * 2 + 1];

        #pragma unroll
        for (int p = 0; p < NP; ++p) {
            const float ox = offp[(l * NP + p) * 2 + 0];
            const float oy = offp[(l * NP + p) * 2 + 1];
            // grid_sample(align_corners=False): x = loc_x*W - 0.5
            const float x = (rx + ox / fw) * fw - 0.5f;
            const float y = (ry + oy / fh) * fh - 0.5f;
            const float x0f = floorf(x);
            const float y0f = floorf(y);
            const float wx = x - x0f;
            const float wy = y - y0f;
            const int x0 = (int)x0f;
            const int y0 = (int)y0f;
            const float aw = la[l * NP + p] * inv;

            #pragma unroll
            for (int c = 0; c < 4; ++c) {
                const int dy = c >> 1;
                const int dx = c & 1;
                const int xi = x0 + dx;
                const int yi = y0 + dy;
                const float cw = (dy ? wy : (1.f - wy)) * (dx ? wx : (1.f - wx));
                // wave-uniform validity test -> uniform branch, loads skipped
                if (xi >= 0 && xi < ww && yi >= 0 && yi < hh) {
                    const float g = memb[(size_t)(base + yi * ww + xi) * CC];
                    acc = fmaf(g, cw * aw, acc);
                }
            }
        }
    }

    attended[(size_t)bq * CC + h * HD + lane] = acc;
}

// ---------------------------------------------------------------------------
// Host-side orchestration
// ---------------------------------------------------------------------------
extern "C" void kernel_launch(void* const* d_in, const int* in_sizes, int n_in,
                              void* d_out, int out_size, void* d_ws, size_t ws_size,
                              hipStream_t stream) {
    const float* query      = (const float*)d_in[0];  // [B,Q,C]
    const float* memory     = (const float*)d_in[1];  // [B,S,C]
    const float* ref_points = (const float*)d_in[2];  // [B,Q,NL,2]
    const float* W_off      = (const float*)d_in[3];  // [C, NH*NL*NP*2 = 256]
    const float* b_off      = (const float*)d_in[4];  // [256]
    const float* W_attn     = (const float*)d_in[5];  // [C, NH*NL*NP = 128]
    const float* b_attn     = (const float*)d_in[6];  // [128]
    const float* W_out      = (const float*)d_in[7];  // [C, C]
    const float* b_out      = (const float*)d_in[8];  // [C]
    float* out = (float*)d_out;

    const int M  = BB * QQ;                    // 4800 rows (multiple of 16)
    constexpr int K  = CC;                     // 256
    constexpr int NO = NH * NL * NP * 2;       // 256 offset columns
    constexpr int NA = NH * NL * NP;           // 128 attention columns

    // workspace layout (floats)
    float* ws_offs = (float*)d_ws;                       // M * NO
    float* ws_attn = ws_offs + (size_t)M * NO;           // M * NA
    float* ws_att  = ws_attn + (size_t)M * NA;           // M * CC (attended)

    const int threads = 128;                   // 4 waves per block
    const int wavesPerBlock = threads / 32;

    // 1) offset projection: [M,K] x [K,256]
    {
        int tiles  = (M / 16) * (NO / 32);
        int blocks = (tiles + wavesPerBlock - 1) / wavesPerBlock;
        gemm_wmma_f32_bias<NO, K><<<blocks, threads, 0, stream>>>(
            query, W_off, b_off, ws_offs, M);
    }
    // 2) attention-logit projection: [M,K] x [K,128]
    {
        int tiles  = (M / 16) * (NA / 32);
        int blocks = (tiles + wavesPerBlock - 1) / wavesPerBlock;
        gemm_wmma_f32_bias<NA, K><<<blocks, threads, 0, stream>>>(
            query, W_attn, b_attn, ws_attn, M);
    }
    // 3) softmax + bilinear gather + weighted aggregation
    {
        int totalWaves = BB * QQ * NH;         // 38400
        int sThreads   = 256;                  // 8 waves per block
        int sWaves     = sThreads / 32;
        int blocks     = (totalWaves + sWaves - 1) / sWaves;
        msda_sample_agg<<<blocks, sThreads, 0, stream>>>(
            memory, ref_points, ws_offs, ws_attn, ws_att);
    }
    // 4) output projection: [M,K] x [K,256] -> d_out
    {
        int tiles  = (M / 16) * (CC / 32);
        int blocks = (tiles + wavesPerBlock - 1) / wavesPerBlock;
        gemm_wmma_f32_bias<CC, K><<<blocks, threads, 0, stream>>>(
            ws_att, W_out, b_out, out, M);
    }
}